// HBV_adj_74921409511426
// MI455X (gfx1250) — compile-verified
//
#include <hip/hip_runtime.h>

// ---------------------------------------------------------------------------
// HBV implicit-Euler integrator, one thread per basin (wave32).
// J = I - d(dS)/dx is block lower-triangular:
//   [J00 J01  .   .   . ]
//   [J10 J11  .   .   . ]
//   [J20 J21 J22  .   . ]
//   [J30 J31 J32 J33  . ]
//   [ .   .   .  J43 J44]
// -> 2x2 Cramer + forward substitution (≈20 flops per solve).
// Climate stream (NT x B x 3 f32) is double-buffered through LDS with CDNA5
// global_load_async_to_lds_b32 (ASYNCcnt) so HBM latency hides behind the
// Newton compute. Each lane consumes only the 12B it staged -> no barriers.
// 64-thread blocks (782 workgroups) maximize WGP spread for the fixed
// 50000-thread budget; Q is written with non-temporal stores to keep the
// 192MB L2 dedicated to the inbound climate stream.
// ---------------------------------------------------------------------------

#define HBV_BLOCK 64
#define NEWTON_ITERS 4

__device__ __forceinline__ float fast_rcp(float x) {
    return __builtin_amdgcn_rcpf(x);
}

// Flat pointer -> wave-relative LDS byte offset (flat LDS addr[31:0] is the
// LDS offset per the CDNA5 aperture rules).
__device__ __forceinline__ unsigned lds_off(const void* p) {
    return (unsigned)(unsigned long long)p;
}

__global__ __launch_bounds__(HBV_BLOCK)
void hbv_implicit_kernel(const float* __restrict__ y0,
                         const float* __restrict__ theta,
                         const float* __restrict__ clim,
                         float* __restrict__ Q,
                         int B, int NT)
{
    __shared__ float sclim[2][HBV_BLOCK * 3];

    const int tid = threadIdx.x;
    const int b   = blockIdx.x * HBV_BLOCK + tid;
    if (b >= B) return;

    const unsigned long long gbase = (unsigned long long)clim;

    // ---- physical parameters: lo + theta * (hi - lo) -----------------------
    const float* th = theta + (size_t)b * 13;
    const float Beta   = 1.0f   + th[0]  * 5.0f;
    const float FC     = 50.0f  + th[1]  * 950.0f;
    const float K0     = 0.05f  + th[2]  * 0.85f;
    const float K1     = 0.01f  + th[3]  * 0.49f;
    const float K2     = 0.001f + th[4]  * 0.199f;
    const float LP     = 0.2f   + th[5]  * 0.8f;
    const float PERC   = 0.0f   + th[6]  * 10.0f;
    const float UZL    = 0.0f   + th[7]  * 100.0f;
    const float TT     = -2.5f  + th[8]  * 5.0f;
    const float CFMAX  = 0.5f   + th[9]  * 9.5f;
    const float CFR    = 0.0f   + th[10] * 0.1f;
    const float CWH    = 0.0f   + th[11] * 0.2f;
    const float BETAET = 0.3f   + th[12] * 4.7f;

    const float invFC   = fast_rcp(FC);
    const float invLPFC = fast_rcp(LP * FC);

    // ---- state -------------------------------------------------------------
    const float* yp = y0 + (size_t)b * 5;
    float p0 = yp[0], p1 = yp[1], p2 = yp[2], p3 = yp[3], p4 = yp[4];

    // ---- async stage of climate[t] slice into LDS buffer `buf` -------------
    auto issue_async = [&](int t, int buf) {
        const unsigned goff =
            (unsigned)(((unsigned long long)t * (unsigned long long)B +
                        (unsigned long long)b) * 12ull);
        const unsigned la = lds_off(&sclim[buf][tid * 3]);
        asm volatile(
            "global_load_async_to_lds_b32 %0, %1, %2\n\t"
            "global_load_async_to_lds_b32 %0, %1, %2 offset:4\n\t"
            "global_load_async_to_lds_b32 %0, %1, %2 offset:8"
            :: "v"(la), "v"(goff), "s"(gbase)
            : "memory");
    };

    issue_async(0, 0);

    for (int t = 0; t < NT; ++t) {
        const int cur = t & 1;
        if (t + 1 < NT) {
            // close the WAR window on buf cur^1 before the async path
            // overwrites it, then kick next day's fetch and wait only for the
            // *older* 3 ops (in-order completion => asynccnt<=3 means day t
            // has landed while day t+1 stays in flight behind compute).
            asm volatile("s_wait_dscnt 0" ::: "memory");
            issue_async(t + 1, cur ^ 1);
            asm volatile("s_wait_asynccnt 3" ::: "memory");
        } else {
            asm volatile("s_wait_asynccnt 0" ::: "memory");
        }
        if (t + 2 < NT) {
            __builtin_prefetch(&clim[((size_t)(t + 2) * B + b) * 3], 0, 1);
        }

        const float P  = sclim[cur][tid * 3 + 0];
        const float T  = sclim[cur][tid * 3 + 1];
        const float Ep = sclim[cur][tid * 3 + 2];

        // state-independent forcings
        const float melt_cap = fmaxf(CFMAX * (T - TT), 0.0f);
        const float refr_cap = fmaxf(CFR * CFMAX * (TT - T), 0.0f);
        const float sf = (T < TT) ? P : 0.0f;
        const float rf = (T < TT) ? 0.0f : P;

        float x0 = p0, x1 = p1, x2 = p2, x3 = p3, x4 = p4;

#pragma unroll
        for (int it = 0; it < NEWTON_ITERS; ++it) {
            // clamps + their indicator derivatives
            const float s0 = fmaxf(x0, 0.0f);   const float c0 = (x0 > 0.0f)  ? 1.0f : 0.0f;
            const float s1 = fmaxf(x1, 0.0f);   const float c1 = (x1 > 0.0f)  ? 1.0f : 0.0f;
            const float s2 = fmaxf(x2, 1e-8f);  const float c2 = (x2 > 1e-8f) ? 1.0f : 0.0f;
            const float s3 = fmaxf(x3, 0.0f);   const float c3 = (x3 > 0.0f)  ? 1.0f : 0.0f;
            const float s4 = fmaxf(x4, 0.0f);   const float c4 = (x4 > 0.0f)  ? 1.0f : 0.0f;
            const float rs2 = fast_rcp(s2);

            // fluxes + partials w.r.t. clamped states
            const bool  refr_on = s1 < refr_cap;
            const float refr = refr_on ? s1 : refr_cap;
            const float dref = refr_on ? 1.0f : 0.0f;

            const bool  melt_on = s0 < melt_cap;
            const float melt = melt_on ? s0 : melt_cap;
            const float dmel = melt_on ? 1.0f : 0.0f;

            const float isn_raw = s1 - CWH * s0;
            const bool  isn_on  = isn_raw > 0.0f;
            const float Isnow = isn_on ? isn_raw : 0.0f;
            const float disn  = isn_on ? 1.0f : 0.0f;   // d/ds1 ; d/ds0 = -CWH*disn
            const float di0   = -CWH * disn;

            const float wraw = __powf(s2 * invFC, Beta);
            const float w    = fminf(wraw, 1.0f);
            const float dw   = (wraw < 1.0f) ? (Beta * wraw * rs2) : 0.0f;

            const float inflow = rf + Isnow;
            const float PEFF   = inflow * w;
            const float dpe2   = inflow * dw;

            const bool  ex_on = s2 > FC;
            const float ex  = ex_on ? (s2 - FC) : 0.0f;
            const float dex = ex_on ? 1.0f : 0.0f;

            const float eraw = __powf(s2 * invLPFC, BETAET);
            const float ef   = fminf(eraw, 1.0f);
            const float def  = (eraw < 1.0f) ? (BETAET * eraw * rs2) : 0.0f;

            const float pet   = Ep * ef;
            const bool  et_sm = s2 < pet;
            const float et    = et_sm ? s2 : pet;
            const float detx  = et_sm ? 1.0f : (Ep * def);

            const bool  perc_on = s3 < PERC;
            const float perc  = perc_on ? s3 : PERC;
            const float dperc = perc_on ? 1.0f : 0.0f;

            const float q0raw = s3 - UZL;
            const bool  q0_on = q0raw > 0.0f;
            const float q0  = q0_on ? (K0 * q0raw) : 0.0f;
            const float dq0 = q0_on ? K0 : 0.0f;

            const float q1 = K1 * s3;
            const float q2 = K2 * s4;

            // residual G = (x - x_prev) - dS   (dt = 1)
            const float g0 = (x0 - p0) - (sf + refr - melt);
            const float g1 = (x1 - p1) - (melt - refr - Isnow);
            const float g2 = (x2 - p2) - (Isnow + rf - PEFF - ex - et);
            const float g3 = (x3 - p3) - (PEFF + ex - perc - q0 - q1);
            const float g4 = (x4 - p4) - (perc - q2);

            // J = I - d(dS)/dx (analytic, sparse)
            const float omw = 1.0f - w;
            const float J00 = 1.0f + dmel * c0;
            const float J01 = -dref * c1;
            const float J10 = -(dmel - di0) * c0;
            const float J11 = 1.0f + (dref + disn) * c1;
            const float J20 = -di0 * omw * c0;
            const float J21 = -disn * omw * c1;
            const float J22 = 1.0f + (dpe2 + dex + detx) * c2;
            const float J30 = -di0 * w * c0;
            const float J31 = -disn * w * c1;
            const float J32 = -(dpe2 + dex) * c2;
            const float J33 = 1.0f + (dperc + dq0 + K1) * c3;
            const float J43 = -dperc * c3;
            const float J44 = 1.0f + K2 * c4;

            // structured solve: 2x2 Cramer then forward substitution
            const float rdet = fast_rcp(J00 * J11 - J01 * J10);
            const float dx0 = (J11 * g0 - J01 * g1) * rdet;
            const float dx1 = (J00 * g1 - J10 * g0) * rdet;
            const float dx2 = (g2 - J20 * dx0 - J21 * dx1) * fast_rcp(J22);
            const float dx3 = (g3 - J30 * dx0 - J31 * dx1 - J32 * dx2) * fast_rcp(J33);
            const float dx4 = (g4 - J43 * dx3) * fast_rcp(J44);

            x0 -= dx0; x1 -= dx1; x2 -= dx2; x3 -= dx3; x4 -= dx4;
        }

        // discharge at the converged state; NT store keeps L2 for the
        // inbound climate stream.
        const float s3f = fmaxf(x3, 0.0f);
        const float s4f = fmaxf(x4, 0.0f);
        const float q = K0 * fmaxf(s3f - UZL, 0.0f) + K1 * s3f + K2 * s4f;
        __builtin_nontemporal_store(q, &Q[(size_t)t * B + b]);

        p0 = x0; p1 = x1; p2 = x2; p3 = x3; p4 = x4;
    }
}

extern "C" void kernel_launch(void* const* d_in, const int* in_sizes, int n_in,
                              void* d_out, int out_size, void* d_ws, size_t ws_size,
                              hipStream_t stream) {
    (void)n_in; (void)d_ws; (void)ws_size;
    const float* y0    = (const float*)d_in[0];   // (B, 5)
    const float* theta = (const float*)d_in[1];   // (B, 13)
    const float* clim  = (const float*)d_in[2];   // (NT, B, 3)
    float* Q = (float*)d_out;                     // (NT, B)

    const int B  = in_sizes[0] / 5;
    const int NT = (B > 0) ? in_sizes[2] / (B * 3) : 0;
    if (B <= 0 || NT <= 0) return;
    (void)out_size;

    const int blocks = (B + HBV_BLOCK - 1) / HBV_BLOCK;
    hipLaunchKernelGGL(hbv_implicit_kernel, dim3(blocks), dim3(HBV_BLOCK), 0,
                       stream, y0, theta, clim, Q, B, NT);
}